// ClassCenterCalculator_40647570489401
// MI455X (gfx1250) — compile-verified
//
#include <hip/hip_runtime.h>

// Problem constants (match reference)
#define BATCH 262144
#define DIM   512
#define NCLS  3

// WMMA kernel tiling
#define GY    64                  // batch chunks in grid.y
#define ROWS  (BATCH / GY)        // 4096 rows per block

typedef __attribute__((ext_vector_type(2))) float v2f;
typedef __attribute__((ext_vector_type(8))) float v8f;

// ---------------------------------------------------------------------------
// Kernel 0: zero the accumulator region of the workspace (harness poisons it)
// ---------------------------------------------------------------------------
__global__ void init_ws(float* __restrict__ sums, unsigned int* __restrict__ counts) {
    int i = blockIdx.x * blockDim.x + threadIdx.x;
    if (i < NCLS * DIM) sums[i] = 0.0f;
    if (i < NCLS) counts[i] = 0u;
}

// ---------------------------------------------------------------------------
// Kernel 1: per-row argmax of pseudo-labels -> predf[] (f32 class id), and
// exact class counts (strict '>' keeps first max, matching jnp.argmax)
// ---------------------------------------------------------------------------
__global__ void argmax_count(const float* __restrict__ labels,
                             float* __restrict__ predf,
                             unsigned int* __restrict__ counts) {
    __shared__ unsigned int hist[NCLS];
    if (threadIdx.x < NCLS) hist[threadIdx.x] = 0u;
    __syncthreads();

    int b = blockIdx.x * blockDim.x + threadIdx.x;   // grid exactly covers BATCH
    float l0 = labels[b * 3 + 0];
    float l1 = labels[b * 3 + 1];
    float l2 = labels[b * 3 + 2];
    int p = 0; float mx = l0;
    if (l1 > mx) { p = 1; mx = l1; }
    if (l2 > mx) { p = 2; }
    predf[b] = (float)p;                             // {0.0f, 1.0f, 2.0f}
    atomicAdd(&hist[p], 1u);

    __syncthreads();
    if (threadIdx.x < NCLS) atomicAdd(&counts[threadIdx.x], hist[threadIdx.x]);
}

// ---------------------------------------------------------------------------
// Kernel 2: segment-sum as GEMM  sums += OneHot(pred)^T x features
// via V_WMMA_F32_16X16X4_F32. Each wave owns one 16-dim column tile and
// consumes 4 batch rows per WMMA; 2 accumulators break the D->C chain.
// Scalar loop control (compile-time trip count), constant instruction
// offsets for all strided loads, f32 pred compare (no byte unpack).
// ---------------------------------------------------------------------------
__global__ void __launch_bounds__(256)
wmma_segsum(const float* __restrict__ features,
            const float* __restrict__ predf,
            float* __restrict__ sums) {
    const int lane = threadIdx.x & 31;
    const int wv   = threadIdx.x >> 5;        // 8 waves per block
    const int m    = lane & 15;               // A row / B,D column within tile
    const int h    = lane >> 4;               // lane half -> K pair {2h, 2h+1}
    const float mf = (float)m;
    const int d0   = blockIdx.x * 16;         // feature-dim tile base

    // This lane's first K row; always even -> predf b64 loads are aligned.
    const int row0 = blockIdx.y * ROWS + wv * 4 + 2 * h;
    const float* f = features + (size_t)row0 * DIM + d0 + m;
    const float* p = predf + row0;

    v8f acc0 = {};
    v8f acc1 = {};

    // 8 waves * 4 rows = 32 rows per sweep; two sweeps per iteration.
    constexpr int ITERS = ROWS / 64;          // 64
    for (int i = 0; i < ITERS; ++i) {
        v2f a0, b0, a1, b1;
        const float pa0 = p[0],  pb0 = p[1];    // merged global_load_b64
        const float pa1 = p[32], pb1 = p[33];
        a0.x = (pa0 == mf) ? 1.0f : 0.0f;       // A[m, 2h]
        a0.y = (pb0 == mf) ? 1.0f : 0.0f;       // A[m, 2h+1]
        a1.x = (pa1 == mf) ? 1.0f : 0.0f;
        a1.y = (pb1 == mf) ? 1.0f : 0.0f;
        b0.x = f[0];                            // B[2h,   m]
        b0.y = f[DIM];                          // B[2h+1, m]
        b1.x = f[32 * DIM];
        b1.y = f[33 * DIM];
        acc0 = __builtin_amdgcn_wmma_f32_16x16x4_f32(
            false, a0, false, b0, (short)0, acc0, false, false);
        acc1 = __builtin_amdgcn_wmma_f32_16x16x4_f32(
            false, a1, false, b1, (short)0, acc1, false, false);
        f += 64 * DIM;
        p += 64;
    }

    v8f acc = acc0 + acc1;

    // D[M = i + 8*h, N = lane%16] : class sums are VGPRs 0..2 on lane half 0
    if (h == 0) {
        atomicAdd(&sums[0 * DIM + d0 + m], acc[0]);
        atomicAdd(&sums[1 * DIM + d0 + m], acc[1]);
        atomicAdd(&sums[2 * DIM + d0 + m], acc[2]);
    }
}

// ---------------------------------------------------------------------------
// Kernel 3: centers = sums / max(count, 1)  (zero-count classes stay zero)
// ---------------------------------------------------------------------------
__global__ void finalize(const float* __restrict__ sums,
                         const unsigned int* __restrict__ counts,
                         float* __restrict__ out) {
    int i = blockIdx.x * blockDim.x + threadIdx.x;
    if (i < NCLS * DIM) {
        unsigned int c = counts[i / DIM];
        float denom = (c > 0u) ? (float)c : 1.0f;
        out[i] = sums[i] / denom;
    }
}

// ---------------------------------------------------------------------------
extern "C" void kernel_launch(void* const* d_in, const int* in_sizes, int n_in,
                              void* d_out, int out_size, void* d_ws, size_t ws_size,
                              hipStream_t stream) {
    (void)in_sizes; (void)n_in; (void)out_size; (void)ws_size;

    const float* features = (const float*)d_in[0];   // [BATCH, DIM] f32
    const float* labels   = (const float*)d_in[1];   // [BATCH, NCLS] f32
    float* out = (float*)d_out;                      // [NCLS, DIM] f32

    // ws layout: [0,6144) f32 sums | [6144,6156) u32 counts | [8192, +1MB) f32 predf
    float*        sums   = (float*)d_ws;
    unsigned int* counts = (unsigned int*)((char*)d_ws + NCLS * DIM * sizeof(float));
    float*        predf  = (float*)((char*)d_ws + 8192);

    init_ws<<<dim3((NCLS * DIM + 255) / 256), dim3(256), 0, stream>>>(sums, counts);
    argmax_count<<<dim3(BATCH / 256), dim3(256), 0, stream>>>(labels, predf, counts);
    wmma_segsum<<<dim3(DIM / 16, GY), dim3(256), 0, stream>>>(features, predf, sums);
    finalize<<<dim3((NCLS * DIM + 255) / 256), dim3(256), 0, stream>>>(sums, counts, out);
}